// Spline_73194832659084
// MI455X (gfx1250) — compile-verified
//
#include <hip/hip_runtime.h>
#include <hip/hip_bf16.h>

typedef __attribute__((ext_vector_type(2))) float v2f;
typedef __attribute__((ext_vector_type(8))) float v8f;

#define HDIM 1024
#define WDIM 1024
// w_wt = ww / (W*W*H*H) = ww * 2^-40
#define W_SCALE 0x1p-40f
#define EPS_R2 1e-10f

// One wave -> 16 consecutive output pixels along w (one row h).
// A-matrix (16x4 f32): weights replicated across all 16 rows.
// B-matrix (4x16 f32): B[k, j] = phi(query j, train k).
// D[i, j] = sum_k w[k]*phi[j,k]  -> identical across rows i == out[query j].
// Per ISA layouts (CDNA5 ISA): lanes 0-15 carry K={0,1} in vgpr{0,1},
// lanes 16-31 carry K={2,3}; so lane computes phi for k = 4*chunk + 2*hi + s.
__global__ __launch_bounds__(256) void spline_tps_wmma_kernel(
    const float* __restrict__ train_points,  // [16,2] in [0,1)
    const float* __restrict__ ww,            // [16]
    const float* __restrict__ vw,            // [3]
    float* __restrict__ out)                 // [4, H, W] f32
{
  const int lane = threadIdx.x & 31;
  const int wv   = threadIdx.x >> 5;
  const int tile = blockIdx.x * 8 + wv;      // 65536 tiles of 16 pixels
  const int h     = tile >> 6;               // W/16 = 64 tiles per row
  const int wbase = (tile & 63) << 4;
  const int j  = lane & 15;                  // query column within tile
  const int hi = lane >> 4;                  // lane half -> K group

  const float qx = (float)(wbase + j);
  const float qy = (float)h;

  // Each lane evaluates its 8 phi values (2 per K-chunk x 4 chunks) and
  // fetches the matching scaled weights for the A matrix.
  float phiv[8];
  float wsc[8];
#pragma unroll
  for (int c = 0; c < 4; ++c) {
#pragma unroll
    for (int s = 0; s < 2; ++s) {
      const int k  = 4 * c + 2 * hi + s;     // global train index this slot maps to
      const float tx = train_points[2 * k]     * (float)WDIM;
      const float ty = train_points[2 * k + 1] * (float)HDIM;
      const float dx = qx - tx;
      const float dy = qy - ty;
      float r2 = fmaf(dx, dx, dy * dy);
      r2 = fmaxf(r2, EPS_R2);
      // r2 in [1e-10, ~2.1e6]: no denorms/negatives -> raw v_sqrt_f32 is safe,
      // avoids the IEEE range-fixup expansion (v_cmp_class/cndmask/rescale).
      phiv[2 * c + s] = r2 * __builtin_amdgcn_sqrtf(r2);  // r^3 = r2^{1.5}
      wsc [2 * c + s] = ww[k] * W_SCALE;
    }
  }

  // K=16 reduction via 4 chained f32 WMMAs.
  v8f acc = {};
#pragma unroll
  for (int c = 0; c < 4; ++c) {
    v2f a; a.x = wsc[2 * c];  a.y = wsc[2 * c + 1];   // A: weights (rows identical)
    v2f b; b.x = phiv[2 * c]; b.y = phiv[2 * c + 1];  // B: phi[query j, k]
    acc = __builtin_amdgcn_wmma_f32_16x16x4_f32(
        /*neg_a=*/false, a, /*neg_b=*/false, b,
        /*c_mod=*/(short)0, acc, /*reuse_a=*/false, /*reuse_b=*/false);
  }

  // Linear term + result. acc[0] holds D[M=0 or 8, N=j]; all rows equal.
  const float lin = fmaf(qx, vw[0], fmaf(qy, vw[1], vw[2]));
  const float val = acc[0] + lin;

  // Batch outputs are identical (inputs broadcast over batch):
  // lanes 0-15 write batches 0,1; lanes 16-31 write batches 2,3.
  const int HW   = HDIM * WDIM;
  const int base = h * WDIM + wbase + j;
  out[(2 * hi)     * HW + base] = val;
  out[(2 * hi + 1) * HW + base] = val;
}

extern "C" void kernel_launch(void* const* d_in, const int* in_sizes, int n_in,
                              void* d_out, int out_size, void* d_ws, size_t ws_size,
                              hipStream_t stream) {
  // d_in[0] = x [4,1024,1024,1] (zeros; only provides batch size -> unused)
  const float* tp = (const float*)d_in[1];   // train_points [1,16,2]
  const float* ww = (const float*)d_in[2];   // ww [1,16,1]
  const float* vw = (const float*)d_in[3];   // vw [1,3,1]
  float* out = (float*)d_out;                // [4,1024,1024,1] f32

  // 1M pixels / 16 per wave = 65536 waves; 8 waves per 256-thread block.
  dim3 grid(8192), block(256);
  hipLaunchKernelGGL(spline_tps_wmma_kernel, grid, block, 0, stream,
                     tp, ww, vw, out);
}